// seq2seq_9234179686975
// MI455X (gfx1250) — compile-verified
//
#include <hip/hip_runtime.h>
#include <hip/hip_bf16.h>
#include <stdint.h>

// ---- problem constants (match reference) ----
#define BB 64      // batch
#define SS 50      // encoder seq len
#define TT 25      // decoder seq len
#define EE 256     // embedding dim
#define UU 512     // units
#define GG 1536    // 3*U
#define VV 32000   // vocab

#define KC 32      // K-chunk staged per TDM transfer (2 x 32x128 f32 = 32KB LDS)

typedef float v2f  __attribute__((ext_vector_type(2)));
typedef float v8f  __attribute__((ext_vector_type(8)));
typedef unsigned int u32x4 __attribute__((ext_vector_type(4)));
typedef int   i32x8 __attribute__((ext_vector_type(8)));
typedef int   i32x4 __attribute__((ext_vector_type(4)));

__device__ __forceinline__ float sigmoidf_(float x) {
    return 1.0f / (1.0f + __expf(-x));
}

// ---------------------------------------------------------------------------
// Issue a TDM DMA: load a KC x 128 f32 tile of B (row-major, leading dim ldb)
// starting at gsrc into LDS at byte offset ldsAddr.  D# per CDNA5 ISA:
//   group0: [count=1][lds_addr][global_addr lo][global_addr hi | type=2]
//   group1: data_size=4B, tensor_dim0=ldb, tensor_dim1=rows, tile=128 x rows,
//           tensor_dim0_stride=ldb
// This toolchain exposes the 6-arg builtin:
//   (u32x4 g0, i32x8 g1, i32x4 g2, i32x4 g3, i32x8 extra, i32 cpol)
// ---------------------------------------------------------------------------
__device__ __forceinline__ void tdm_load_tile(const float* gsrc, int ldb,
                                              uint32_t ldsAddr, int rows, int cols)
{
    uint64_t ga = (uint64_t)(uintptr_t)gsrc;
    u32x4 g0;
    g0[0] = 1u;                                              // count=1, user mode
    g0[1] = ldsAddr;                                         // lds_addr (bytes)
    g0[2] = (uint32_t)ga;                                    // global_addr[31:0]
    g0[3] = (uint32_t)((ga >> 32) & 0x01FFFFFFu) | (2u << 30); // addr[56:32] | type=2
    i32x8 g1;
    g1[0] = (2 << 16);                                       // data_size=2 (4 bytes)
    g1[1] = (ldb & 0xFFFF) << 16;                            // tensor_dim0[15:0]
    g1[2] = ((ldb >> 16) & 0xFFFF) | ((rows & 0xFFFF) << 16); // dim0 hi | tensor_dim1 lo
    g1[3] = (cols & 0xFFFF) << 16;                           // tensor_dim1 hi(0) | tile_dim0
    g1[4] = rows & 0xFFFF;                                   // tile_dim1 | tile_dim2=0
    g1[5] = ldb;                                             // tensor_dim0_stride[31:0]
    g1[6] = 0;                                               // stride hi | dim1_stride lo
    g1[7] = 0;
    i32x4 z4 = {0, 0, 0, 0};
    i32x8 z8 = {0, 0, 0, 0, 0, 0, 0, 0};
    __builtin_amdgcn_tensor_load_to_lds(g0, g1, z4, z4, z8, 0);
}

// ---------------------------------------------------------------------------
// f32 WMMA GEMM with TDM-staged, double-buffered B tiles in LDS.
// C[M,N] = A[M,K] @ B[K,N] (+ bias[N]) (+ addm[M,N])
// Workgroup (8 waves) computes a 64(M) x 128(N) tile; wave (mtl, nh) computes
// the 16x64 strip at rows mtl*16, cols nh*64 using V_WMMA_F32_16X16X4_F32.
// A-fragment (ISA 7.12.2, 32-bit A 16x4): lane m = lane&15, K pair at 2*(lane>>4).
// Requires M%64==0, N%128==0, K%KC==0.
// ---------------------------------------------------------------------------
__global__ void __launch_bounds__(256)
gemm_wmma_f32(const float* __restrict__ A, int lda,
              const float* __restrict__ Bm, int ldb,
              float* __restrict__ C, int ldc,
              const float* __restrict__ bias,
              const float* __restrict__ addm, int ldadd,
              int nBlocks, int K)
{
    __shared__ float Bs[2][KC * 128];

    int blk = blockIdx.x;
    int mb  = blk / nBlocks;
    int nb  = blk - mb * nBlocks;

    int tid  = threadIdx.x;
    int wv   = tid >> 5;
    int lane = tid & 31;
    int mtl  = wv & 3;           // local m-tile (0..3)
    int nh   = wv >> 2;          // n half (0..1)
    int mn   = lane & 15;
    int kb   = (lane >> 4) << 1; // K sub-offset for this half-wave
    int half = lane >> 4;

    const float* Arow = A + (long)(mb * 64 + mtl * 16 + mn) * lda;
    const float* Bbase = Bm + (long)nb * 128;
    int localn0 = nh * 64 + mn;  // column within the 128-wide LDS tile

    int nCh = K / KC;

    if (wv == 0) {
        tdm_load_tile(Bbase, ldb, (uint32_t)(uintptr_t)&Bs[0][0], KC, 128);
    }

    v8f acc0 = {}, acc1 = {}, acc2 = {}, acc3 = {};

    for (int ic = 0; ic < nCh; ++ic) {
        if (wv == 0) {
            if (ic + 1 < nCh) {
                tdm_load_tile(Bbase + (long)(ic + 1) * KC * ldb, ldb,
                              (uint32_t)(uintptr_t)&Bs[(ic + 1) & 1][0], KC, 128);
                __builtin_amdgcn_s_wait_tensorcnt(1);   // oldest (chunk ic) done
            } else {
                __builtin_amdgcn_s_wait_tensorcnt(0);
            }
        }
        __syncthreads();                                 // LDS tile ic visible to all

        const float* Bsb = &Bs[ic & 1][0];
        const float* Ak  = Arow + ic * KC;
#pragma unroll
        for (int k0 = 0; k0 < KC; k0 += 4) {
            v2f a = *(const v2f*)(Ak + k0 + kb);         // A[m][k+kb], A[m][k+kb+1]
            const float* r0 = Bsb + (k0 + kb) * 128;
            const float* r1 = r0 + 128;
            v2f b0, b1, b2, b3;
            b0.x = r0[localn0];      b0.y = r1[localn0];
            b1.x = r0[localn0 + 16]; b1.y = r1[localn0 + 16];
            b2.x = r0[localn0 + 32]; b2.y = r1[localn0 + 32];
            b3.x = r0[localn0 + 48]; b3.y = r1[localn0 + 48];
            acc0 = __builtin_amdgcn_wmma_f32_16x16x4_f32(false, a, false, b0, (short)0, acc0, false, false);
            acc1 = __builtin_amdgcn_wmma_f32_16x16x4_f32(false, a, false, b1, (short)0, acc1, false, false);
            acc2 = __builtin_amdgcn_wmma_f32_16x16x4_f32(false, a, false, b2, (short)0, acc2, false, false);
            acc3 = __builtin_amdgcn_wmma_f32_16x16x4_f32(false, a, false, b3, (short)0, acc3, false, false);
        }
        __syncthreads();                                 // done reading buffer
    }

    v8f accs[4] = {acc0, acc1, acc2, acc3};
#pragma unroll
    for (int j = 0; j < 4; ++j) {
        int col = nb * 128 + nh * 64 + j * 16 + mn;
        float bv = bias ? bias[col] : 0.0f;
#pragma unroll
        for (int i = 0; i < 8; ++i) {
            int row = mb * 64 + mtl * 16 + i + half * 8;
            float v = accs[j][i] + bv;
            if (addm) v += addm[(long)row * ldadd + col];
            C[(long)row * ldc + col] = v;
        }
    }
}

// ---- embedding gather: out[r, 0:E] = table[ids[r], 0:E] ----
__global__ void gather_rows(const float* __restrict__ table,
                            const int* __restrict__ ids,
                            float* __restrict__ out, int rows)
{
    long idx = (long)blockIdx.x * blockDim.x + threadIdx.x;
    long total = (long)rows * EE;
    if (idx >= total) return;
    int r = (int)(idx / EE);
    int c = (int)(idx - (long)r * EE);
    out[idx] = table[(long)ids[r] * EE + c];
}

__global__ void fill_zero(float* p, int n)
{
    int i = blockIdx.x * blockDim.x + threadIdx.x;
    if (i < n) p[i] = 0.0f;
}

// ---- encoder GRU pointwise (reset_after=True) ----
__global__ void gru_enc_pointwise(const float* __restrict__ XW,
                                  const float* __restrict__ hU,
                                  float* __restrict__ h,
                                  float* __restrict__ enc_out, int t)
{
    int idx = blockIdx.x * blockDim.x + threadIdx.x;   // B*U
    if (idx >= BB * UU) return;
    int b = idx >> 9;
    int u = idx & (UU - 1);
    const float* xrow = XW + (long)(b * SS + t) * GG;
    const float* hrow = hU + (long)b * GG;
    float z  = sigmoidf_(xrow[u] + hrow[u]);
    float r  = sigmoidf_(xrow[UU + u] + hrow[UU + u]);
    float hh = tanhf(xrow[2 * UU + u] + r * hrow[2 * UU + u]);
    float hn = z * h[idx] + (1.0f - z) * hh;
    h[idx] = hn;
    enc_out[(long)(b * SS + t) * UU + u] = hn;
}

// ---- Bahdanau attention + context, one workgroup per batch element ----
__global__ void __launch_bounds__(256)
attention_kernel(const float* __restrict__ enc_proj,
                 const float* __restrict__ enc_out,
                 const float* __restrict__ attnW,
                 const float* __restrict__ Vw,
                 const float* __restrict__ bV,
                 float* __restrict__ ctx)
{
    int b = blockIdx.x;
    __shared__ float sc[SS];
    int tid = threadIdx.x, lane = tid & 31, wv = tid >> 5;
    const float* aw_b = attnW + (long)b * UU;

    for (int s = wv; s < SS; s += 8) {
        const float* ep = enc_proj + (long)(b * SS + s) * UU;
        float p = 0.0f;
        for (int u = lane; u < UU; u += 32)
            p += tanhf(ep[u] + aw_b[u]) * Vw[u];
        for (int off = 16; off; off >>= 1) p += __shfl_xor(p, off, 32);
        if (lane == 0) sc[s] = p + bV[0];
    }
    __syncthreads();

    if (tid < 32) {                           // softmax over S=50 in one wave
        float mx = -3.4e38f;
        for (int s = lane; s < SS; s += 32) mx = fmaxf(mx, sc[s]);
        for (int off = 16; off; off >>= 1) mx = fmaxf(mx, __shfl_xor(mx, off, 32));
        float sum = 0.0f;
        for (int s = lane; s < SS; s += 32) { float e = __expf(sc[s] - mx); sc[s] = e; sum += e; }
        for (int off = 16; off; off >>= 1) sum += __shfl_xor(sum, off, 32);
        float inv = 1.0f / sum;
        for (int s = lane; s < SS; s += 32) sc[s] *= inv;
    }
    __syncthreads();

    const float* eo_b = enc_out + (long)b * SS * UU;
    for (int u = tid; u < UU; u += 256) {
        float acc = 0.0f;
        for (int s = 0; s < SS; ++s) acc += sc[s] * eo_b[(long)s * UU + u];
        ctx[(long)b * UU + u] = acc;
    }
}

// ---- decoder GRU pointwise: h_prev = 0 => state = (1-z)*hh, hz = b_d[1] ----
__global__ void gru_dec_pointwise(const float* __restrict__ gz,
                                  const float* __restrict__ b_d1,   // b_d + G
                                  float* __restrict__ h,
                                  float* __restrict__ states, int t)
{
    int idx = blockIdx.x * blockDim.x + threadIdx.x;   // B*U
    if (idx >= BB * UU) return;
    int b = idx >> 9;
    int u = idx & (UU - 1);
    const float* grow = gz + (long)b * GG;
    float z  = sigmoidf_(grow[u] + b_d1[u]);
    float r  = sigmoidf_(grow[UU + u] + b_d1[UU + u]);
    float hh = tanhf(grow[2 * UU + u] + r * b_d1[2 * UU + u]);
    float st = (1.0f - z) * hh;
    h[idx] = st;
    states[((long)b * TT + t) * UU + u] = st;
}

// -------------------------------------------------------------------------
extern "C" void kernel_launch(void* const* d_in, const int* in_sizes, int n_in,
                              void* d_out, int out_size, void* d_ws, size_t ws_size,
                              hipStream_t stream)
{
    const int*   inp  = (const int*)d_in[0];
    const int*   tar  = (const int*)d_in[1];
    const float* Eenc = (const float*)d_in[2];
    const float* Wx_e = (const float*)d_in[3];
    const float* Uh_e = (const float*)d_in[4];
    const float* b_e  = (const float*)d_in[5];   // [2,G]
    const float* Edec = (const float*)d_in[6];
    const float* Wx_d = (const float*)d_in[7];   // [(E+U),G], first U rows multiply ctx
    // d_in[8] = Uh_d: unused (decoder GRU starts from zeros -> hz = b_d[1])
    const float* b_d  = (const float*)d_in[9];   // [2,G]
    const float* W1   = (const float*)d_in[10];
    const float* b1   = (const float*)d_in[11];
    const float* W2   = (const float*)d_in[12];
    const float* b2   = (const float*)d_in[13];
    const float* Vw   = (const float*)d_in[14];
    const float* bV   = (const float*)d_in[15];
    const float* Wfc  = (const float*)d_in[16];
    const float* bfc  = (const float*)d_in[17];
    float* out = (float*)d_out;                  // [B,T,V]

    // workspace layout (floats)
    float* ws = (float*)d_ws;
    float* x_emb    = ws;                         // 3200*256
    float* dec_emb  = x_emb    + (long)BB*SS*EE;  // 1600*256
    float* XW       = dec_emb  + (long)BB*TT*EE;  // 3200*1536
    float* enc_out  = XW       + (long)BB*SS*GG;  // 3200*512
    float* enc_proj = enc_out  + (long)BB*SS*UU;  // 3200*512
    float* EMBW     = enc_proj + (long)BB*SS*UU;  // 1600*1536
    float* h        = EMBW     + (long)BB*TT*GG;  // 64*512
    float* hU       = h        + (long)BB*UU;     // 64*1536
    float* attnW    = hU       + (long)BB*GG;     // 64*512
    float* ctx      = attnW    + (long)BB*UU;     // 64*512
    float* gz       = ctx      + (long)BB*UU;     // 64*1536
    float* states   = gz       + (long)BB*GG;     // 1600*512

    auto gemm = [&](const float* A, int lda, const float* Bm, int ldb,
                    float* C, int ldc, const float* bias,
                    const float* addm, int ldadd, int M, int N, int K) {
        int mBlocks = M / 64, nBlocks = N / 128;
        gemm_wmma_f32<<<mBlocks * nBlocks, 256, 0, stream>>>(A, lda, Bm, ldb, C, ldc,
                                                             bias, addm, ldadd,
                                                             nBlocks, K);
    };

    // --- embeddings ---
    {
        long n1 = (long)BB * SS * EE;
        gather_rows<<<(int)((n1 + 255) / 256), 256, 0, stream>>>(Eenc, inp, x_emb, BB * SS);
        long n2 = (long)BB * TT * EE;
        gather_rows<<<(int)((n2 + 255) / 256), 256, 0, stream>>>(Edec, tar, dec_emb, BB * TT);
    }

    // --- h0 = 0 ---
    fill_zero<<<(BB * UU + 255) / 256, 256, 0, stream>>>(h, BB * UU);

    // --- XW = x_emb @ Wx_e + b_e[0]   (batched over all (b,s)) ---
    gemm(x_emb, EE, Wx_e, GG, XW, GG, b_e, nullptr, 0, BB * SS, GG, EE);

    // --- encoder recurrence ---
    for (int t = 0; t < SS; ++t) {
        gemm(h, UU, Uh_e, GG, hU, GG, b_e + GG, nullptr, 0, BB, GG, UU);
        gru_enc_pointwise<<<(BB * UU + 255) / 256, 256, 0, stream>>>(XW, hU, h, enc_out, t);
    }
    // h now holds enc_hidden (decoder's initial attention query)

    // --- enc_proj = enc_output @ W1 + b1 (hoisted) ---
    gemm(enc_out, UU, W1, UU, enc_proj, UU, b1, nullptr, 0, BB * SS, UU, UU);

    // --- EMBW = dec_emb @ Wx_d[U:] + b_d[0] (hoisted; embedding half of xcat GEMM) ---
    gemm(dec_emb, EE, Wx_d + (long)UU * GG, GG, EMBW, GG, b_d, nullptr, 0, BB * TT, GG, EE);

    // --- decoder loop (sequential only through attention query) ---
    for (int t = 0; t < TT; ++t) {
        gemm(h, UU, W2, UU, attnW, UU, b2, nullptr, 0, BB, UU, UU);
        attention_kernel<<<BB, 256, 0, stream>>>(enc_proj, enc_out, attnW, Vw, bV, ctx);
        // gz = ctx @ Wx_d[:U] + EMBW[:,t,:]   (b_d[0] already folded into EMBW)
        gemm(ctx, UU, Wx_d, GG, gz, GG, nullptr, EMBW + (long)t * GG, TT * GG, BB, GG, UU);
        gru_dec_pointwise<<<(BB * UU + 255) / 256, 256, 0, stream>>>(gz, b_d + GG, h, states, t);
    }

    // --- logits = states @ Wfc + bfc  (single batched 1600x512x32000 GEMM) ---
    gemm(states, UU, Wfc, VV, out, VV, bfc, nullptr, 0, BB * TT, VV, UU);
}